// RoiAlign_56221121904938
// MI455X (gfx1250) — compile-verified
//
#include <hip/hip_runtime.h>
#include <stdint.h>

typedef __attribute__((ext_vector_type(2))) float v2f;
typedef __attribute__((ext_vector_type(8))) float v8f;

#define NANCH_MAX 196416
#define TOPK 500
#define NC   80
#define CAND_CAP 4096

// ---- workspace layout (bytes) ----
#define OFF_SCORES 0u            // 196416 f32
#define OFF_HIST1  786432u       // 2048 u32
#define OFF_HIST2  794624u       // 4096 u32
#define OFF_SEL    811008u       // 16 u32: [0]=b1 [1]=above1 [2]=b2 [3]=T [4]=cand_cnt
#define OFF_CKEY   811072u       // 4096 u32
#define OFF_CIDX   827456u       // 4096 u32
#define OFF_NBOX   843840u       // 500*4 f32 (y1/h, x1/w, y2/h, x2/w)
#define OFF_TIDX   851840u       // 500 i32
#define ZERO_BYTES (2048u*4u + 4096u*4u + 64u)   // hist1+hist2+sel (contiguous)

// -------- 1) score = max over 80 classes --------
__global__ void k_scores(const float* __restrict__ cls, float* __restrict__ scores, int n) {
  int a = blockIdx.x * blockDim.x + threadIdx.x;
  if (a >= n) return;
  const float4* row = (const float4*)(cls + (size_t)a * NC);
  float m = -1e30f;
#pragma unroll
  for (int j = 0; j < NC / 4; ++j) {
    float4 v = row[j];
    m = fmaxf(m, fmaxf(fmaxf(v.x, v.y), fmaxf(v.z, v.w)));
  }
  scores[a] = m;
}

// -------- 2) radix-select histograms --------
__global__ void k_hist1(const float* __restrict__ scores, unsigned* __restrict__ hist1, int n) {
  int a = blockIdx.x * blockDim.x + threadIdx.x;
  if (a >= n) return;
  unsigned k = __float_as_uint(scores[a]);   // all scores positive -> order preserving
  atomicAdd(&hist1[k >> 20], 1u);
}

__global__ void k_sel1(const unsigned* __restrict__ hist1, unsigned* __restrict__ sel) {
  if (threadIdx.x != 0) return;
  unsigned above = 0;
  for (int b = 2047; b >= 0; --b) {
    unsigned c = hist1[b];
    if (above + c >= TOPK) { sel[0] = (unsigned)b; sel[1] = above; return; }
    above += c;
  }
  sel[0] = 0; sel[1] = above;
}

__global__ void k_hist2(const float* __restrict__ scores, const unsigned* __restrict__ sel,
                        unsigned* __restrict__ hist2, int n) {
  int a = blockIdx.x * blockDim.x + threadIdx.x;
  if (a >= n) return;
  unsigned k = __float_as_uint(scores[a]);
  if ((k >> 20) == sel[0]) atomicAdd(&hist2[(k >> 8) & 0xFFFu], 1u);
}

__global__ void k_sel2(const unsigned* __restrict__ hist2, unsigned* __restrict__ sel) {
  if (threadIdx.x != 0) return;
  unsigned above = sel[1];
  unsigned b1 = sel[0];
  for (int b = 4095; b >= 0; --b) {
    unsigned c = hist2[b];
    if (above + c >= TOPK) { sel[2] = (unsigned)b; sel[3] = (b1 << 20) | ((unsigned)b << 8); return; }
    above += c;
  }
  sel[2] = 0; sel[3] = (b1 << 20);
}

__global__ void k_compact(const float* __restrict__ scores, unsigned* __restrict__ sel,
                          unsigned* __restrict__ ckey, unsigned* __restrict__ cidx, int n) {
  int a = blockIdx.x * blockDim.x + threadIdx.x;
  if (a >= n) return;
  unsigned k = __float_as_uint(scores[a]);
  if (k >= sel[3]) {
    unsigned p = atomicAdd(&sel[4], 1u);
    if (p < CAND_CAP) { ckey[p] = k; cidx[p] = (unsigned)a; }
  }
}

// -------- 3) one-block bitonic sort of candidates + gather outputs --------
__global__ __launch_bounds__(1024) void k_final(
    const unsigned* __restrict__ ckey, const unsigned* __restrict__ cidx,
    const unsigned* __restrict__ sel, const float* __restrict__ boxes,
    const float* __restrict__ cls, const int* __restrict__ ishape,
    float* __restrict__ out, float* __restrict__ nbox, int* __restrict__ tidx) {
  __shared__ unsigned long long keys[CAND_CAP];
  __shared__ int sidx[TOPK];
  int tid = threadIdx.x;
  unsigned cnt = sel[4]; if (cnt > CAND_CAP) cnt = CAND_CAP;
  for (int i = tid; i < CAND_CAP; i += 1024)
    keys[i] = (i < (int)cnt)
                ? (((unsigned long long)ckey[i] << 32) | (unsigned)(~cidx[i]))  // ties -> lower idx first
                : 0ull;
  __syncthreads();
  for (unsigned k = 2; k <= CAND_CAP; k <<= 1) {
    for (unsigned j = k >> 1; j > 0; j >>= 1) {
      for (unsigned t = tid; t < CAND_CAP / 2; t += 1024) {
        unsigned i = 2u * t - (t & (j - 1u));
        unsigned p = i | j;
        bool desc = ((i & k) == 0u);
        unsigned long long A = keys[i], B = keys[p];
        if ((A < B) == desc) { keys[i] = B; keys[p] = A; }
      }
      __syncthreads();
    }
  }
  float h = (float)ishape[1];
  float w = (float)ishape[2];
  if (tid < TOPK) {
    unsigned long long kk = keys[tid];
    int a = (int)(~(unsigned)(kk & 0xFFFFFFFFull));
    sidx[tid] = a;
    tidx[tid] = a;
    float bx1 = boxes[(size_t)a * 4 + 0], by1 = boxes[(size_t)a * 4 + 1];
    float bx2 = boxes[(size_t)a * 4 + 2], by2 = boxes[(size_t)a * 4 + 3];
    out[tid * 4 + 0] = bx1; out[tid * 4 + 1] = by1;
    out[tid * 4 + 2] = bx2; out[tid * 4 + 3] = by2;
    nbox[tid * 4 + 0] = by1 / h; nbox[tid * 4 + 1] = bx1 / w;
    nbox[tid * 4 + 2] = by2 / h; nbox[tid * 4 + 3] = bx2 / w;
  }
  __syncthreads();
  const float4* c4 = (const float4*)cls;
  float4* o4 = (float4*)(out + TOPK * 4);
  for (int q = tid; q < TOPK * (NC / 4); q += 1024) {
    int r = q / (NC / 4), j = q % (NC / 4);
    o4[q] = c4[(size_t)sidx[r] * (NC / 4) + j];
  }
}

// -------- 4) crop_and_resize: separable bilinear; vertical pass as FP32 WMMA --------
// out[iy,(ix,c)] = sum_k Wy[iy,k] * hb[k,(ix,c)],  K = 28 (y0/y1 per output row),
// x-blend (+x-mask) folded into hb, y-weights (+y-mask) in Wy. Exact f32 math.
__global__ __launch_bounds__(256) void k_roi(
    const float* __restrict__ p3, const float* __restrict__ p4, const float* __restrict__ p5,
    const float* __restrict__ p6, const float* __restrict__ p7,
    const float* __restrict__ nbox, float* __restrict__ rout) {
  int lvl = blockIdx.x;   // 0..4
  int box = blockIdx.y;   // 0..499
  int tid = threadIdx.x;
  const float* fp = (lvl == 0) ? p3 : (lvl == 1) ? p4 : (lvl == 2) ? p5 : (lvl == 3) ? p6 : p7;
  int H = 128 >> lvl;     // square feature map, 256 channels

  __shared__ float s_wy[16], s_wx[16], s_my[16], s_mx[16];
  __shared__ int   s_y0[16], s_y1[16], s_x0[16], s_x1[16];
  __shared__ float hb[28][14][16];   // x-blended rows for current 16-channel slab
  __shared__ float ob[14][14][16];   // WMMA output staging for coalesced stores

  if (tid < 32) {
    bool isY = tid < 16;
    int i = tid & 15;
    float n1 = nbox[box * 4 + (isY ? 0 : 1)];
    float n2 = nbox[box * 4 + (isY ? 2 : 3)];
    float t  = (float)i * (1.0f / 13.0f);
    float s  = (n1 + t * (n2 - n1)) * (float)(H - 1);
    float f0 = floorf(s);
    float wq = s - f0;
    int  i0  = (int)f0;
    int  c0  = min(max(i0, 0), H - 1);
    int  c1  = min(max(i0 + 1, 0), H - 1);
    float mk = (s >= 0.0f && s <= (float)(H - 1)) ? 1.0f : 0.0f;
    if (i >= 14) { wq = 0.0f; c0 = 0; c1 = 0; mk = 0.0f; }
    if (isY) { s_wy[i] = wq; s_y0[i] = c0; s_y1[i] = c1; s_my[i] = mk; }
    else     { s_wx[i] = wq; s_x0[i] = c0; s_x1[i] = c1; s_mx[i] = mk; }
  }
  __syncthreads();

  int lane = tid & 31;
  int lh   = lane >> 4;          // lane half: selects K pair per ISA A/B layout
  int wvid = tid >> 5;           // wave id (wave32)
  size_t obase = (size_t)box * (14 * 14 * 1280) + (size_t)lvl * 256;

  for (int cp = 0; cp < 16; ++cp) {        // 16 slabs of 16 channels
    int cbase = cp * 16;
    // ---- horizontal pass: gather + x-blend into LDS ----
    for (int idx = tid; idx < 28 * 14 * 16; idx += 256) {
      int c  = idx & 15;
      int ix = (idx >> 4) % 14;
      int k  = idx / 224;
      int iy = k >> 1;
      int yr = (k & 1) ? s_y1[iy] : s_y0[iy];
      const float* rowp = fp + (size_t)yr * H * 256;
      float v0  = rowp[(size_t)s_x0[ix] * 256 + cbase + c];
      float v1  = rowp[(size_t)s_x1[ix] * 256 + cbase + c];
      float wxv = s_wx[ix];
      hb[k][ix][c] = s_mx[ix] * ((1.0f - wxv) * v0 + wxv * v1);
    }
    __syncthreads();
    // ---- vertical pass: V_WMMA_F32_16X16X4_F32, K=28 in 7 chunks ----
    for (int ix = wvid; ix < 14; ix += 8) {     // wave-uniform => EXEC stays all-ones
      int m = lane & 15;   // A: M = lane%16
      int c = lane & 15;   // B/D: N = lane%16
      float wyv = s_wy[m];
      float myv = s_my[m];
      v8f acc = {};
#pragma unroll
      for (int kc = 0; kc < 7; ++kc) {
        int k0 = kc * 4 + lh * 2;             // lanes 0-15: K{0,1}; lanes 16-31: K{2,3}
        v2f a, b;
        a.x = (k0     == 2 * m) ? (1.0f - wyv) * myv : ((k0     == 2 * m + 1) ? wyv * myv : 0.0f);
        a.y = (k0 + 1 == 2 * m) ? (1.0f - wyv) * myv : ((k0 + 1 == 2 * m + 1) ? wyv * myv : 0.0f);
        b.x = hb[k0][ix][c];
        b.y = hb[k0 + 1][ix][c];
        acc = __builtin_amdgcn_wmma_f32_16x16x4_f32(false, a, false, b, (short)0, acc,
                                                    false, false);
      }
#pragma unroll
      for (int i = 0; i < 8; ++i) {
        int mr = i + lh * 8;                  // D: VGPR i -> M=i (lanes 0-15) / i+8 (16-31)
        if (mr < 14) ob[mr][ix][c] = acc[i];
      }
    }
    __syncthreads();
    // ---- coalesced float4 stores of this slab ----
    const float4* ob4 = (const float4*)&ob[0][0][0];
    for (int q = tid; q < 784; q += 256) {    // 14*14*16/4
      int pix = q >> 2;
      int c4i = q & 3;
      *(float4*)(rout + obase + (size_t)pix * 1280 + cbase + c4i * 4) = ob4[q];
    }
    __syncthreads();
  }
}

extern "C" void kernel_launch(void* const* d_in, const int* in_sizes, int n_in,
                              void* d_out, int out_size, void* d_ws, size_t ws_size,
                              hipStream_t stream) {
  const int*   ishape = (const int*)d_in[0];
  const float* boxes  = (const float*)d_in[1];
  const float* cls    = (const float*)d_in[2];
  const float* p3     = (const float*)d_in[3];
  const float* p4     = (const float*)d_in[4];
  const float* p5     = (const float*)d_in[5];
  const float* p6     = (const float*)d_in[6];
  const float* p7     = (const float*)d_in[7];
  float* out = (float*)d_out;
  char*  ws  = (char*)d_ws;

  float*    scores = (float*)(ws + OFF_SCORES);
  unsigned* hist1  = (unsigned*)(ws + OFF_HIST1);
  unsigned* hist2  = (unsigned*)(ws + OFF_HIST2);
  unsigned* sel    = (unsigned*)(ws + OFF_SEL);
  unsigned* ckey   = (unsigned*)(ws + OFF_CKEY);
  unsigned* cidx   = (unsigned*)(ws + OFF_CIDX);
  float*    nbox   = (float*)(ws + OFF_NBOX);
  int*      tidx   = (int*)(ws + OFF_TIDX);

  int n = in_sizes[1] / 4;   // 196416 anchors
  int blocks = (n + 255) / 256;

  // re-zero histograms / selection state / candidate counter every call (graph-safe)
  hipMemsetAsync(ws + OFF_HIST1, 0, ZERO_BYTES, stream);

  k_scores <<<blocks, 256, 0, stream>>>(cls, scores, n);
  k_hist1  <<<blocks, 256, 0, stream>>>(scores, hist1, n);
  k_sel1   <<<1, 32, 0, stream>>>(hist1, sel);
  k_hist2  <<<blocks, 256, 0, stream>>>(scores, sel, hist2, n);
  k_sel2   <<<1, 32, 0, stream>>>(hist2, sel);
  k_compact<<<blocks, 256, 0, stream>>>(scores, sel, ckey, cidx, n);
  k_final  <<<1, 1024, 0, stream>>>(ckey, cidx, sel, boxes, cls, ishape, out, nbox, tidx);
  k_roi    <<<dim3(5, 500), 256, 0, stream>>>(p3, p4, p5, p6, p7, nbox,
                                              out + TOPK * 4 + TOPK * NC);
}